// MoeMuxExpertChoiceAllTokens_12498354831791
// MI455X (gfx1250) — compile-verified
//
#include <hip/hip_runtime.h>
#include <math.h>

// Problem constants (match reference): B,T,D,H,O,E
#define Bc 32
#define Tc 2048
#define Dc 1024
#define Hc 4096
#define Oc 1024
#define Ec 16

typedef __attribute__((ext_vector_type(2))) float v2f;
typedef __attribute__((ext_vector_type(8))) float v8f;

// CDNA5 f32 WMMA: D(16x16,f32) = A(16x4,f32) * B(4x16,f32) + C
__device__ __forceinline__ v8f wmma4(v2f a, v2f b, v8f c) {
  return __builtin_amdgcn_wmma_f32_16x16x4_f32(false, a, false, b, (short)0, c,
                                               false, false);
}

// Wave32 operand lane mapping (ISA 7.12.2):
//   A 16x4 (MxK):  lane l holds row m=l&15; vgpr j holds k = j + 2*(l>=16)
//   B 4x16 (KxN):  lane l holds col n=l&15; vgpr j holds k = j + 2*(l>=16)
//   C/D 16x16:     lane l holds col n=l&15; vgpr r holds row m = r + 8*(l>=16)

// ---------------------------------------------------------------------------
// K1: gate logits  logits[b,t,e] = x[b,t,:] . gate_w[e,:] + gate_b[e]
// One wave per 32-token pair of tiles; the gate_w B-fragment is loaded once
// and feeds two independent WMMA accumulation chains.
// ---------------------------------------------------------------------------
__global__ void gate_kernel(const float* __restrict__ x,
                            const float* __restrict__ gw,
                            const float* __restrict__ gb,
                            float* __restrict__ logits) {
  const int wave = blockIdx.x * (blockDim.x >> 5) + (threadIdx.x >> 5);
  const int lane = threadIdx.x & 31;
  const int half = lane >> 4;
  const int l16  = lane & 15;
  const long tok0 = (long)wave * 32;            // flat token index (b*T + t)
  const float* xr0 = x + (tok0 + l16) * (long)Dc;
  const float* xr1 = x + (tok0 + 16 + l16) * (long)Dc;
  const float* gwp = gw + (long)l16 * Dc;       // B col (expert n = l16)
  v8f c0 = {}, c1 = {};
  for (int k0 = 0; k0 < Dc; k0 += 4) {
    const int ka = k0 + 2 * half;
    v2f b  = {gwp[ka], gwp[ka + 1]};            // shared B fragment
    v2f a0 = {xr0[ka], xr0[ka + 1]};
    v2f a1 = {xr1[ka], xr1[ka + 1]};
    c0 = wmma4(a0, b, c0);
    c1 = wmma4(a1, b, c1);
  }
  const float bias = gb[l16];
  float* op = logits + tok0 * Ec;
#pragma unroll
  for (int r = 0; r < 8; ++r) {
    const int row = r + 8 * half;
    op[(long)row * Ec + l16]        = c0[r] + bias;
    op[(long)(16 + row) * Ec + l16] = c1[r] + bias;
  }
}

// ---------------------------------------------------------------------------
// K2: per-(b,e) stats for softmax over tokens (axis T)
// ---------------------------------------------------------------------------
__global__ void tok_stats_kernel(const float* __restrict__ logits,
                                 float* __restrict__ tmax,
                                 float* __restrict__ tsum) {
  const int be = blockIdx.x;              // b*E + e
  const int b = be / Ec, e = be % Ec;
  const float* base = logits + ((long)b * Tc) * Ec + e;
  __shared__ float red[256];
  float m = -1e30f;
  for (int t = threadIdx.x; t < Tc; t += blockDim.x)
    m = fmaxf(m, base[(long)t * Ec]);
  red[threadIdx.x] = m;
  __syncthreads();
  for (int s = 128; s > 0; s >>= 1) {
    if (threadIdx.x < s)
      red[threadIdx.x] = fmaxf(red[threadIdx.x], red[threadIdx.x + s]);
    __syncthreads();
  }
  m = red[0];
  __syncthreads();
  float sum = 0.f;
  for (int t = threadIdx.x; t < Tc; t += blockDim.x)
    sum += expf(base[(long)t * Ec] - m);
  red[threadIdx.x] = sum;
  __syncthreads();
  for (int s = 128; s > 0; s >>= 1) {
    if (threadIdx.x < s) red[threadIdx.x] += red[threadIdx.x + s];
    __syncthreads();
  }
  if (threadIdx.x == 0) { tmax[be] = m; tsum[be] = red[0]; }
}

// ---------------------------------------------------------------------------
// K3: per-token probabilities: p_exp (softmax over E), p_tok (softmax over T)
// ---------------------------------------------------------------------------
__global__ void probs_kernel(const float* __restrict__ logits,
                             const float* __restrict__ tmax,
                             const float* __restrict__ tsum,
                             float* __restrict__ p_tok,
                             float* __restrict__ p_exp) {
  const long tok = (long)blockIdx.x * blockDim.x + threadIdx.x;
  if (tok >= (long)Bc * Tc) return;
  const int b = (int)(tok / Tc);
  const float* l = logits + tok * Ec;
  float v[Ec], ex[Ec];
  float m = -1e30f;
#pragma unroll
  for (int e = 0; e < Ec; ++e) { v[e] = l[e]; m = fmaxf(m, v[e]); }
  float s = 0.f;
#pragma unroll
  for (int e = 0; e < Ec; ++e) { ex[e] = expf(v[e] - m); s += ex[e]; }
  const float inv = 1.f / s;
  float* pe = p_exp + tok * Ec;
  float* pt = p_tok + tok * Ec;
#pragma unroll
  for (int e = 0; e < Ec; ++e) {
    pe[e] = ex[e] * inv;
    pt[e] = expf(v[e] - tmax[b * Ec + e]) / tsum[b * Ec + e];
  }
}

// ---------------------------------------------------------------------------
// K4: soft dispatch  inp[e,b,d] = sum_t p_tok[b,t,e] * x[b,t,d]
// per b: (E x T) @ (T x D). One wave per 32-wide D slab: the p_tok A-fragment
// is loaded once per k-step and feeds two D-tiles.
// ---------------------------------------------------------------------------
__global__ void dispatch_kernel(const float* __restrict__ x,
                                const float* __restrict__ p_tok,
                                float* __restrict__ inp) {
  const int wave = blockIdx.x * (blockDim.x >> 5) + (threadIdx.x >> 5);
  const int lane = threadIdx.x & 31;
  const int half = lane >> 4;
  const int l16  = lane & 15;
  const int b  = wave >> 5;          // 32 d-slabs per batch
  const int d0 = (wave & 31) * 32;
  const float* pt = p_tok + (long)b * Tc * Ec;   // A[m=e][k=t] = pt[t*E + e]
  const float* xb = x + (long)b * Tc * Dc + d0 + l16;  // B[k=t][n]
  v8f c0 = {}, c1 = {};
  for (int t0 = 0; t0 < Tc; t0 += 4) {
    const int ta = t0 + 2 * half;
    v2f a  = {pt[(long)ta * Ec + l16], pt[(long)(ta + 1) * Ec + l16]};
    v2f b0 = {xb[(long)ta * Dc],      xb[(long)(ta + 1) * Dc]};
    v2f b1 = {xb[(long)ta * Dc + 16], xb[(long)(ta + 1) * Dc + 16]};
    c0 = wmma4(a, b0, c0);
    c1 = wmma4(a, b1, c1);
  }
#pragma unroll
  for (int r = 0; r < 8; ++r) {
    const int e = r + 8 * half;      // D row m = expert
    inp[((long)e * Bc + b) * Dc + d0 + l16]      = c0[r];
    inp[((long)e * Bc + b) * Dc + d0 + 16 + l16] = c1[r];
  }
}

// ---------------------------------------------------------------------------
// K5: expert FFN layer 1 + exact GELU
// h[e,b,:] = gelu( inp[e,b,:] @ w1[e,0:D,:] + w1[e,D,:] )
// One wave computes the FULL M=32 x 32-wide N slab: each weight fragment is
// fetched exactly once (halves the dominant HBM stream) and feeds 4
// independent WMMA chains.
// ---------------------------------------------------------------------------
__global__ void ffn1_kernel(const float* __restrict__ inp,
                            const float* __restrict__ w1,
                            float* __restrict__ h) {
  const int wave = blockIdx.x * (blockDim.x >> 5) + (threadIdx.x >> 5);
  const int lane = threadIdx.x & 31;
  const int half = lane >> 4;
  const int l16  = lane & 15;
  const int e  = wave >> 7;                 // 128 n-slabs per expert
  const int h0 = (wave & 127) * 32;
  const float* A0 = inp + ((long)e * Bc + l16) * Dc;       // b rows 0..15
  const float* A1 = A0 + (long)16 * Dc;                    // b rows 16..31
  const float* Bp = w1 + (long)e * (Dc + 1) * Hc + h0 + l16;
  v8f c00 = {}, c01 = {}, c10 = {}, c11 = {};
  for (int k0 = 0; k0 < Dc; k0 += 4) {
    if ((k0 & 31) == 0)
      __builtin_prefetch(Bp + (long)(k0 + 64) * Hc, 0, 1);
    const int ka = k0 + 2 * half;
    v2f a0 = {A0[ka], A0[ka + 1]};
    v2f a1 = {A1[ka], A1[ka + 1]};
    v2f b0 = {Bp[(long)ka * Hc],      Bp[(long)(ka + 1) * Hc]};
    v2f b1 = {Bp[(long)ka * Hc + 16], Bp[(long)(ka + 1) * Hc + 16]};
    c00 = wmma4(a0, b0, c00);
    c01 = wmma4(a0, b1, c01);
    c10 = wmma4(a1, b0, c10);
    c11 = wmma4(a1, b1, c11);
  }
  const float bias0 = Bp[(long)Dc * Hc];        // row D of w1 = bias
  const float bias1 = Bp[(long)Dc * Hc + 16];
  float* hp = h + (long)e * Bc * Hc + h0 + l16;
#pragma unroll
  for (int r = 0; r < 8; ++r) {
    const int row = r + 8 * half;
    float v00 = c00[r] + bias0, v01 = c01[r] + bias1;
    float v10 = c10[r] + bias0, v11 = c11[r] + bias1;
    v00 = 0.5f * v00 * (1.f + erff(v00 * 0.70710678118654752f));
    v01 = 0.5f * v01 * (1.f + erff(v01 * 0.70710678118654752f));
    v10 = 0.5f * v10 * (1.f + erff(v10 * 0.70710678118654752f));
    v11 = 0.5f * v11 * (1.f + erff(v11 * 0.70710678118654752f));
    hp[(long)row * Hc]             = v00;
    hp[(long)row * Hc + 16]        = v01;
    hp[(long)(16 + row) * Hc]      = v10;
    hp[(long)(16 + row) * Hc + 16] = v11;
  }
}

// ---------------------------------------------------------------------------
// K6: expert FFN layer 2 (same 2M x 2N blocking as K5)
// out_e[e,b,:] = h[e,b,:] @ w2[e,0:H,:] + w2[e,H,:]
// ---------------------------------------------------------------------------
__global__ void ffn2_kernel(const float* __restrict__ h,
                            const float* __restrict__ w2,
                            float* __restrict__ oute) {
  const int wave = blockIdx.x * (blockDim.x >> 5) + (threadIdx.x >> 5);
  const int lane = threadIdx.x & 31;
  const int half = lane >> 4;
  const int l16  = lane & 15;
  const int e  = wave >> 5;                 // 32 n-slabs per expert
  const int o0 = (wave & 31) * 32;
  const float* A0 = h + ((long)e * Bc + l16) * Hc;
  const float* A1 = A0 + (long)16 * Hc;
  const float* Bp = w2 + (long)e * (Hc + 1) * Oc + o0 + l16;
  v8f c00 = {}, c01 = {}, c10 = {}, c11 = {};
  for (int k0 = 0; k0 < Hc; k0 += 4) {
    if ((k0 & 31) == 0)
      __builtin_prefetch(Bp + (long)(k0 + 64) * Oc, 0, 1);
    const int ka = k0 + 2 * half;
    v2f a0 = {A0[ka], A0[ka + 1]};
    v2f a1 = {A1[ka], A1[ka + 1]};
    v2f b0 = {Bp[(long)ka * Oc],      Bp[(long)(ka + 1) * Oc]};
    v2f b1 = {Bp[(long)ka * Oc + 16], Bp[(long)(ka + 1) * Oc + 16]};
    c00 = wmma4(a0, b0, c00);
    c01 = wmma4(a0, b1, c01);
    c10 = wmma4(a1, b0, c10);
    c11 = wmma4(a1, b1, c11);
  }
  const float bias0 = Bp[(long)Hc * Oc];
  const float bias1 = Bp[(long)Hc * Oc + 16];
  float* op = oute + (long)e * Bc * Oc + o0 + l16;
#pragma unroll
  for (int r = 0; r < 8; ++r) {
    const int row = r + 8 * half;
    op[(long)row * Oc]             = c00[r] + bias0;
    op[(long)row * Oc + 16]        = c01[r] + bias1;
    op[(long)(16 + row) * Oc]      = c10[r] + bias0;
    op[(long)(16 + row) * Oc + 16] = c11[r] + bias1;
  }
}

// ---------------------------------------------------------------------------
// K7: soft combine  out[b,t,o] = sum_e p_exp[b,t,e] * out_e[e,b,o]
// per b: (T x E) @ (E x O), K = 16. One wave per 32-token pair of tiles; the
// out_e B-fragment is loaded once per k-step and feeds two T-tiles.
// ---------------------------------------------------------------------------
__global__ void combine_kernel(const float* __restrict__ p_exp,
                               const float* __restrict__ oute,
                               float* __restrict__ out) {
  const int wave = blockIdx.x * (blockDim.x >> 5) + (threadIdx.x >> 5);
  const int lane = threadIdx.x & 31;
  const int half = lane >> 4;
  const int l16  = lane & 15;
  const int b  = wave >> 12;                // 64 t-pairs * 64 o-tiles per b
  const int t0 = ((wave >> 6) & 63) * 32;
  const int o0 = (wave & 63) * 16;
  const float* A0 = p_exp + ((long)b * Tc + t0 + l16) * Ec;  // A[m=t][k=e]
  const float* A1 = A0 + (long)16 * Ec;
  const float* Bp = oute + (long)b * Oc + o0 + l16;  // B[k] = Bp[k*B*O]
  v8f c0 = {}, c1 = {};
#pragma unroll
  for (int k0 = 0; k0 < Ec; k0 += 4) {
    const int ka = k0 + 2 * half;
    v2f bb = {Bp[(long)ka * Bc * Oc], Bp[(long)(ka + 1) * Bc * Oc]};
    v2f a0 = {A0[ka], A0[ka + 1]};
    v2f a1 = {A1[ka], A1[ka + 1]};
    c0 = wmma4(a0, bb, c0);
    c1 = wmma4(a1, bb, c1);
  }
  float* op = out + ((long)b * Tc + t0) * Oc + o0 + l16;
#pragma unroll
  for (int r = 0; r < 8; ++r) {
    const int row = r + 8 * half;
    op[(long)row * Oc]        = c0[r];
    op[(long)(16 + row) * Oc] = c1[r];
  }
}

// ---------------------------------------------------------------------------
extern "C" void kernel_launch(void* const* d_in, const int* in_sizes, int n_in,
                              void* d_out, int out_size, void* d_ws,
                              size_t ws_size, hipStream_t stream) {
  (void)in_sizes; (void)n_in; (void)out_size; (void)ws_size;
  const float* x  = (const float*)d_in[0];
  const float* gw = (const float*)d_in[1];
  const float* gb = (const float*)d_in[2];
  const float* w1 = (const float*)d_in[3];
  const float* w2 = (const float*)d_in[4];
  float* out = (float*)d_out;

  // Workspace layout (floats), total ~25.2 MB
  float* ws = (float*)d_ws;
  size_t off = 0;
  float* logits = ws + off; off += (size_t)Bc * Tc * Ec;   // 4 MB
  float* tmax   = ws + off; off += (size_t)Bc * Ec;
  float* tsum   = ws + off; off += (size_t)Bc * Ec;
  float* p_tok  = ws + off; off += (size_t)Bc * Tc * Ec;   // 4 MB
  float* p_exp  = ws + off; off += (size_t)Bc * Tc * Ec;   // 4 MB
  float* inp    = ws + off; off += (size_t)Ec * Bc * Dc;   // 2 MB
  float* hbuf   = ws + off; off += (size_t)Ec * Bc * Hc;   // 8 MB
  float* oute   = ws + off; off += (size_t)Ec * Bc * Oc;   // 2 MB

  // 8 waves (256 threads) per block.
  gate_kernel<<<(Bc * Tc / 32) / 8, 256, 0, stream>>>(x, gw, gb, logits);
  tok_stats_kernel<<<Bc * Ec, 256, 0, stream>>>(logits, tmax, tsum);
  probs_kernel<<<(Bc * Tc) / 256, 256, 0, stream>>>(logits, tmax, tsum,
                                                    p_tok, p_exp);
  dispatch_kernel<<<(Bc * (Dc / 32)) / 8, 256, 0, stream>>>(x, p_tok, inp);
  ffn1_kernel<<<(Ec * (Hc / 32)) / 8, 256, 0, stream>>>(inp, w1, hbuf);
  ffn2_kernel<<<(Ec * (Oc / 32)) / 8, 256, 0, stream>>>(hbuf, w2, oute);
  combine_kernel<<<(Bc * (Tc / 32) * (Oc / 16)) / 8, 256, 0, stream>>>(
      p_exp, oute, out);
}